// NormDropout_56401510531541
// MI455X (gfx1250) — compile-verified
//
#include <hip/hip_runtime.h>
#include <hip/hip_bf16.h>

// Problem constants (from the reference)
#define C_DIM      1024
#define H_DIM      256
#define W_DIM      256
#define FH         128
#define FW         128
#define L_DIM      16384          // FH*FW
#define NUM_DROP   819            // int(16384*0.1 // 2)
#define NUM_LEFT   15565          // 16384 - 819
#define OUT_DROP   1638           // NUM_DROP*2
#define OUT_PER_C  63898          // 1638 + 15565*4

typedef __attribute__((ext_vector_type(2))) float v2f;
typedef __attribute__((ext_vector_type(8))) float v8f;

// ---------------------------------------------------------------------------
// Kernel 1: norm^2 per location via V_WMMA_F32_16X16X4_F32 Gram diagonal.
// One wave per 16-location tile (tiles never cross a row of the 128x128 grid).
// Lane mapping matches the WMMA 32-bit A layout: lane&15 = M (location),
// lanes 0-15 hold K=0(vx)/K=1(vy), lanes 16-31 hold K=2(vx)/K=3(vy).
// With B = A (Gram), D[m][m] accumulates sum_k m_val[m][k]^2 over all chunks.
// ---------------------------------------------------------------------------
__global__ __launch_bounds__(256) void nd_norms_kernel(
    const float* __restrict__ x, float* __restrict__ norms) {
  const int lane = threadIdx.x & 31;
  const int wave = threadIdx.x >> 5;
  const int tile = blockIdx.x * 8 + wave;          // 0..1023
  const int fy   = tile >> 3;                      // 8 tiles per feature row
  const int fx   = (tile & 7) * 16 + (lane & 15);  // this lane's location col
  const int cofs = (lane < 16) ? 0 : 2;            // K base for this half-wave
  const size_t rowBase = (size_t)(fy * 2) * W_DIM + (size_t)fx * 2;

  v8f acc = {};
  for (int k0 = 0; k0 < C_DIM; k0 += 4) {
    const int c0 = k0 + cofs;
    const float* p0 = x + ((size_t)c0 << 16) + rowBase;        // channel c0
    const float* p1 = x + ((size_t)(c0 + 1) << 16) + rowBase;  // channel c0+1
    float2 a00 = *(const float2*)(p0);
    float2 a01 = *(const float2*)(p0 + W_DIM);
    float2 a10 = *(const float2*)(p1);
    float2 a11 = *(const float2*)(p1 + W_DIM);
    v2f a;
    a.x = 0.25f * ((a00.x + a00.y) + (a01.x + a01.y));  // m[loc][c0]
    a.y = 0.25f * ((a10.x + a10.y) + (a11.x + a11.y));  // m[loc][c0+1]
    // D = A * A^T + C ; diagonal accumulates per-location sum of squares
    acc = __builtin_amdgcn_wmma_f32_16x16x4_f32(
        false, a, false, a, (short)0, acc, false, false);
  }

  // Extract diagonal: C/D layout -> VGPR j holds M=j (lanes 0-15, N=lane)
  // and M=j+8 (lanes 16-31, N=lane-16). Diagonal M==N:
  //   loc=j    lives in acc[j] of lane j        (j=0..7)
  //   loc=j+8  lives in acc[j] of lane j+24
  int loc = -1;
  float v = 0.0f;
#pragma unroll
  for (int j = 0; j < 8; ++j) {
    if (lane == j)      { v = acc[j]; loc = j; }
    if (lane == j + 24) { v = acc[j]; loc = j + 8; }
  }
  if (loc >= 0) norms[tile * 16 + loc] = v;   // norm^2 (sqrt is monotonic)
}

// ---------------------------------------------------------------------------
// Kernel 2: single-workgroup bitonic argsort of 16384 (key,idx) pairs in LDS.
// 128 KB of LDS -- only possible thanks to CDNA5's 320 KB-per-WGP LDS.
// Index tie-break reproduces jnp.argsort's stable order.
// ---------------------------------------------------------------------------
__global__ __launch_bounds__(1024) void nd_sort_kernel(
    const float* __restrict__ norms, int* __restrict__ order) {
  __shared__ float skey[L_DIM];
  __shared__ int   sidx[L_DIM];
  for (int i = threadIdx.x; i < L_DIM; i += 1024) {
    skey[i] = norms[i];
    sidx[i] = i;
  }
  __syncthreads();
  for (int k = 2; k <= L_DIM; k <<= 1) {
    for (int j = k >> 1; j > 0; j >>= 1) {
      for (int i = threadIdx.x; i < L_DIM; i += 1024) {
        const int ixj = i ^ j;
        if (ixj > i) {
          const bool up = ((i & k) == 0);
          float ka = skey[i], kb = skey[ixj];
          int   ia = sidx[i], ib = sidx[ixj];
          const bool agtb = (ka > kb) || (ka == kb && ia > ib);
          if (up ? agtb : !agtb) {
            skey[i] = kb; skey[ixj] = ka;
            sidx[i] = ib; sidx[ixj] = ia;
          }
        }
      }
      __syncthreads();
    }
  }
  for (int i = threadIdx.x; i < L_DIM; i += 1024) order[i] = sidx[i];
}

// ---------------------------------------------------------------------------
// Kernel 3: replicate jax.random.uniform(key=42, (819,4)) via threefry2x32
// and take the top-2 tap indices per dropped location (stable descending).
// ---------------------------------------------------------------------------
__device__ __forceinline__ unsigned nd_rotl(unsigned v, int n) {
  return (v << n) | (v >> (32 - n));
}
__device__ void nd_threefry2x32(unsigned k0, unsigned k1,
                                unsigned x0, unsigned x1,
                                unsigned& o0, unsigned& o1) {
  const unsigned ks0 = k0, ks1 = k1, ks2 = k0 ^ k1 ^ 0x1BD11BDAu;
  x0 += ks0; x1 += ks1;
  const int r0[4] = {13, 15, 26, 6};
  const int r1[4] = {17, 29, 16, 24};
#pragma unroll
  for (int g = 1; g <= 5; ++g) {
    const int* rr = (g & 1) ? r0 : r1;
#pragma unroll
    for (int r = 0; r < 4; ++r) {
      x0 += x1; x1 = nd_rotl(x1, rr[r]); x1 ^= x0;
    }
    const unsigned ks[3] = {ks0, ks1, ks2};
    x0 += ks[g % 3];
    x1 += ks[(g + 1) % 3] + (unsigned)g;
  }
  o0 = x0; o1 = x1;
}

__global__ __launch_bounds__(1024) void nd_scores_kernel(int* __restrict__ sind) {
  const int j = threadIdx.x;
  if (j >= NUM_DROP) return;
  float sc[4];
#pragma unroll
  for (int c = 0; c < 4; ++c) {
    const unsigned i = (unsigned)(j * 4 + c);     // flat index into (819,4)
    unsigned o0, o1, bits;
    // JAX splits iota(3276) in halves: bits[i<1638] = out0(i, i+1638),
    // bits[i>=1638] = out1(i-1638, i). key(42) -> (0, 42).
    if (i < 1638u) { nd_threefry2x32(0u, 42u, i, i + 1638u, o0, o1); bits = o0; }
    else           { nd_threefry2x32(0u, 42u, i - 1638u, i, o0, o1); bits = o1; }
    unsigned u = (bits >> 9) | 0x3F800000u;
    sc[c] = __uint_as_float(u) - 1.0f;            // uniform [0,1)
  }
  // top-2, stable (first occurrence wins ties), matching argsort(-scores)
  int s0 = 0;
#pragma unroll
  for (int c = 1; c < 4; ++c) if (sc[c] > sc[s0]) s0 = c;
  int s1 = -1;
#pragma unroll
  for (int c = 0; c < 4; ++c) {
    if (c == s0) continue;
    if (s1 < 0 || sc[c] > sc[s1]) s1 = c;
  }
  sind[j]            = s0;   // sample_ind[0][j]
  sind[NUM_DROP + j] = s1;   // sample_ind[1][j]
}

// ---------------------------------------------------------------------------
// Kernel 4: gather + relayout. Coalesced writes; gathers hit L2 because the
// per-channel working set (256 KB) is tiny vs the 192 MB L2 and blocks are
// scheduled channel-major (gridDim.y = channel).
// ---------------------------------------------------------------------------
__global__ __launch_bounds__(256) void nd_gather_kernel(
    const float* __restrict__ x, const int* __restrict__ order,
    const int* __restrict__ sind, float* __restrict__ out) {
  const int p = blockIdx.x * 256 + threadIdx.x;
  if (p >= OUT_PER_C) return;
  const int c = blockIdx.y;

  int l, e;
  if (p < OUT_DROP) {
    const int j = p >> 1;
    const int s = p & 1;
    l = order[j];                    // idx_drop[j]
    e = sind[s * NUM_DROP + j];      // sample_ind[s][j]
  } else {
    const int q = p - OUT_DROP;
    l = order[NUM_DROP + (q >> 2)];  // idx_left[k]
    e = q & 3;
  }
  const int fy = l >> 7, fx = l & 127;
  const int yy = fy * 2 + (e >> 1);
  const int xx = fx * 2 + (e & 1);
  out[(size_t)c * OUT_PER_C + p] = x[((size_t)c << 16) + (size_t)yy * W_DIM + xx];
}

// ---------------------------------------------------------------------------
extern "C" void kernel_launch(void* const* d_in, const int* in_sizes, int n_in,
                              void* d_out, int out_size, void* d_ws, size_t ws_size,
                              hipStream_t stream) {
  const float* x = (const float*)d_in[0];
  float* out = (float*)d_out;

  float* norms = (float*)d_ws;                           // 16384 f32 (64 KB)
  int*   order = (int*)((char*)d_ws + 64 * 1024);        // 16384 i32 (64 KB)
  int*   sind  = (int*)((char*)d_ws + 128 * 1024);       // 2*819 i32

  // Phase 1: per-location channel norms via WMMA (1024 tiles, 8 waves/block)
  nd_norms_kernel<<<128, 256, 0, stream>>>(x, norms);
  // Phase 2: full argsort of 16384 norms in 128 KB of LDS (one WGP)
  nd_sort_kernel<<<1, 1024, 0, stream>>>(norms, order);
  // Phase 3: threefry scores + top-2 tap selection
  nd_scores_kernel<<<1, 1024, 0, stream>>>(sind);
  // Phase 4: gather/relayout, channel-major
  nd_gather_kernel<<<dim3((OUT_PER_C + 255) / 256, C_DIM), 256, 0, stream>>>(
      x, order, sind, out);
}